// MultiEncoderDR_34643206209546
// MI455X (gfx1250) — compile-verified
//
#include <hip/hip_runtime.h>

typedef __attribute__((ext_vector_type(2))) float v2f;
typedef __attribute__((ext_vector_type(8))) float v8f;

#define QB      8      // qubits
#define NSTATE  256    // 2^QB
#define LAYERS  2
#define ENC     16
#define DIM     512
#define BTILE   16

__launch_bounds__(256)
__global__ void qenc_fused(const float* __restrict__ x,
                           const float* __restrict__ proj_w,
                           const float* __restrict__ qw,
                           float* __restrict__ out)
{
    __shared__ float  ampsS[BTILE * NSTATE];   // 16KB: amps tile, later Z-reduction scratch
    __shared__ float2 st[BTILE * NSTATE];      // 32KB: 16 statevectors (complex)
    __shared__ float  red[BTILE][16];
    __shared__ float  invn[BTILE];

    const int t     = threadIdx.x;             // 0..255 (8 waves of 32)
    const int lane  = t & 31;
    const int wave  = t >> 5;
    const int bt    = blockIdx.x;              // batch tile
    const int e     = blockIdx.y;              // encoder
    const int b0    = bt * BTILE;

    const int mlane = lane & 15;               // A-frag row / B-frag col / C-frag col
    const int khalf = lane >> 4;               // K-half select

    const float* xrow = x + (size_t)(b0 + mlane) * DIM + 2 * khalf;
    const float* pwE  = proj_w + (size_t)e * NSTATE * DIM;

    // ---------------- GEMM phase: amps[16][256] = x_tile * proj_w[e]^T ------------
    // wave w covers N-tiles {w, w+8}; fp32 WMMA 16x16x4, K = 512 -> 128 WMMA/tile
    #pragma unroll
    for (int half = 0; half < 2; ++half) {
        const int n0 = (wave + half * 8) * 16;
        const float* brow = pwE + (size_t)(n0 + mlane) * DIM + 2 * khalf;
        v8f acc = {0.f, 0.f, 0.f, 0.f, 0.f, 0.f, 0.f, 0.f};
        for (int k0 = 0; k0 < DIM; k0 += 4) {
            v2f a = *(const v2f*)(xrow + k0);   // A[m][k0 + 2*khalf + {0,1}]
            v2f b = *(const v2f*)(brow + k0);   // B[k0 + 2*khalf + {0,1}][n]
            acc = __builtin_amdgcn_wmma_f32_16x16x4_f32(
                      false, a, false, b, (short)0, acc, false, false);
        }
        #pragma unroll
        for (int r = 0; r < 8; ++r)            // C: vgpr r -> row r + 8*khalf
            ampsS[(r + 8 * khalf) * NSTATE + n0 + mlane] = acc[r];
    }
    __syncthreads();

    // ---------------- Normalize + init statevectors --------------------------------
    const int row = t >> 4;                    // 16 threads per batch row
    const int li  = t & 15;
    {
        float ss = 0.f;
        #pragma unroll 4
        for (int j = 0; j < 16; ++j) {
            float v = ampsS[row * NSTATE + li * 16 + j];
            ss += v * v;
        }
        red[row][li] = ss;
    }
    __syncthreads();
    if (t < BTILE) {
        float s = 0.f;
        #pragma unroll
        for (int j = 0; j < 16; ++j) s += red[t][j];
        invn[t] = rsqrtf(s);
    }
    __syncthreads();
    {
        float inv = invn[row];
        #pragma unroll 4
        for (int j = 0; j < 16; ++j) {
            int s = li * 16 + j;
            st[row * NSTATE + s] = make_float2(ampsS[row * NSTATE + s] * inv, 0.f);
        }
    }
    __syncthreads();

    // ---------------- StronglyEntanglingLayers --------------------------------------
    const float* wE = qw + (size_t)e * (LAYERS * QB * 3);
    for (int l = 0; l < LAYERS; ++l) {
        // Rot(phi,theta,omega) = RZ(omega) RY(theta) RZ(phi) on each qubit
        for (int q = 0; q < QB; ++q) {
            float phi   = wE[(l * QB + q) * 3 + 0];
            float theta = wE[(l * QB + q) * 3 + 1];
            float omega = wE[(l * QB + q) * 3 + 2];
            float sh, ch; __sincosf(0.5f * theta,          &sh, &ch);
            float sp, cp; __sincosf(0.5f * (phi + omega),  &sp, &cp);
            float sm, cm; __sincosf(0.5f * (phi - omega),  &sm, &cm);
            // U00 = e^{-i(phi+omega)/2} c ; U01 = -e^{+i(phi-omega)/2} s
            // U10 = e^{-i(phi-omega)/2} s ; U11 = e^{+i(phi+omega)/2} c
            const float2 U00 = make_float2( cp * ch, -sp * ch);
            const float2 U01 = make_float2(-cm * sh, -sm * sh);
            const float2 U10 = make_float2( cm * sh, -sm * sh);
            const float2 U11 = make_float2( cp * ch,  sp * ch);

            const int bit  = (QB - 1) - q;     // wire 0 = MSB
            const int mask = 1 << bit;
            const int lowm = mask - 1;
            #pragma unroll
            for (int p8 = 0; p8 < 8; ++p8) {   // 2048 pairs / 256 threads
                int p  = t + p8 * 256;
                int sv = p >> 7;
                int j  = p & 127;
                int i0 = ((j & ~lowm) << 1) | (j & lowm);
                int i1 = i0 | mask;
                float2 a0 = st[sv * NSTATE + i0];
                float2 a1 = st[sv * NSTATE + i1];
                float2 r0, r1;
                r0.x = U00.x*a0.x - U00.y*a0.y + U01.x*a1.x - U01.y*a1.y;
                r0.y = U00.x*a0.y + U00.y*a0.x + U01.x*a1.y + U01.y*a1.x;
                r1.x = U10.x*a0.x - U10.y*a0.y + U11.x*a1.x - U11.y*a1.y;
                r1.y = U10.x*a0.y + U10.y*a0.x + U11.x*a1.y + U11.y*a1.x;
                st[sv * NSTATE + i0] = r0;
                st[sv * NSTATE + i1] = r1;
            }
            __syncthreads();
        }
        // CNOT cascade: control q -> target (q+r) % QB, r = (l % (QB-1)) + 1
        const int rr = (l % (QB - 1)) + 1;
        for (int q = 0; q < QB; ++q) {
            const int tq  = (q + rr) & (QB - 1);
            const int bc  = (QB - 1) - q;
            const int btb = (QB - 1) - tq;
            const int lo  = bc < btb ? bc : btb;
            const int hi  = bc < btb ? btb : bc;
            const int lom = (1 << lo) - 1;
            const int him = (1 << hi) - 1;
            #pragma unroll
            for (int p4 = 0; p4 < 4; ++p4) {   // 1024 swap-pairs / 256 threads
                int p  = t + p4 * 256;
                int sv = p >> 6;
                int j  = p & 63;
                int i  = ((j & ~lom) << 1) | (j & lom);   // insert 0 at lo
                i      = ((i & ~him) << 1) | (i & him);   // insert 0 at hi
                i     |= (1 << bc);                        // control = 1, target = 0
                int i2 = i | (1 << btb);
                float2 v0 = st[sv * NSTATE + i];
                float2 v1 = st[sv * NSTATE + i2];
                st[sv * NSTATE + i]  = v1;
                st[sv * NSTATE + i2] = v0;
            }
            __syncthreads();
        }
    }

    // ---------------- <Z_q> expectation values --------------------------------------
    float z[QB];
    #pragma unroll
    for (int q = 0; q < QB; ++q) z[q] = 0.f;
    for (int j = 0; j < 16; ++j) {
        int s = li * 16 + j;
        float2 a = st[row * NSTATE + s];
        float p = a.x * a.x + a.y * a.y;
        #pragma unroll
        for (int q = 0; q < QB; ++q)
            z[q] += ((s >> ((QB - 1) - q)) & 1) ? -p : p;
    }
    #pragma unroll
    for (int q = 0; q < QB; ++q)
        ampsS[row * NSTATE + li * 16 + q] = z[q];   // reuse amps LDS as scratch
    __syncthreads();
    if (t < BTILE * QB) {                           // 128 outputs per workgroup
        int orow = t >> 3, oq = t & 7;
        float s = 0.f;
        #pragma unroll
        for (int j = 0; j < 16; ++j)
            s += ampsS[orow * NSTATE + j * 16 + oq];
        out[(size_t)(b0 + orow) * (ENC * QB) + e * QB + oq] = s;
    }
}

extern "C" void kernel_launch(void* const* d_in, const int* in_sizes, int n_in,
                              void* d_out, int out_size, void* d_ws, size_t ws_size,
                              hipStream_t stream) {
    const float* x  = (const float*)d_in[0];   // [B, 512]
    const float* pw = (const float*)d_in[1];   // [16, 256, 512]
    const float* qw = (const float*)d_in[2];   // [16, 2, 8, 3]
    float* out = (float*)d_out;                // [B, 128]
    const int B = in_sizes[0] / DIM;           // 2048
    dim3 grid(B / BTILE, ENC);
    qenc_fused<<<grid, 256, 0, stream>>>(x, pw, qw, out);
}